// CapsuleLayer_63651415327450
// MI455X (gfx1250) — compile-verified
//
#include <hip/hip_runtime.h>
#include <math.h>

// CapsuleLayer dynamic routing, fused (never materializes the 419MB priors).
// One workgroup = (capsule c, 16-batch tile). 8 waves, wave32, WMMA f32 16x16x4.
// M = o (16 output dims), N = b (16 batches), K = i (Cin=8 -> two k4 steps).

typedef float v2f __attribute__((ext_vector_type(2)));
typedef float v8f __attribute__((ext_vector_type(8)));

#define B_    64
#define CIN   8
#define R_    800
#define NC_   128
#define O_    16
#define BT    16          // batch tile = WMMA N
#define NWAVE 8
#define RPW   (R_ / NWAVE)   // 100 r-values per wave

__device__ __forceinline__ v8f wmma_f32_16x16x4(v2f a, v2f b, v8f c) {
  // 8-arg pattern: (neg_a, A, neg_b, B, c_mod, C, reuse_a, reuse_b)
  return __builtin_amdgcn_wmma_f32_16x16x4_f32(false, a, false, b, (short)0, c,
                                               false, false);
}

__global__ __launch_bounds__(256) void capsule_route_kernel(
    const float* __restrict__ x,     // [B][Cin][R]
    const float* __restrict__ w,     // [NC][R][Cin][O]
    float* __restrict__ out)         // [B][O][NC]
{
  __shared__ float logits[BT][R_ + 1];        // +1 pad: kill bank conflicts
  __shared__ float wacc[NWAVE][BT][O_];
  __shared__ float s_lds[BT][O_];
  __shared__ float v_lds[BT][O_];
  __shared__ float pmax[16][BT];
  __shared__ float psum[16][BT];
  __shared__ float mxl[BT];
  __shared__ float rsl[BT];

  const int t    = threadIdx.x;
  const int bt   = blockIdx.x;      // 0..3  (batch tile)
  const int c    = blockIdx.y;      // 0..127 (capsule)
  const int lane = t & 31;
  const int wv   = t >> 5;
  const int mn   = lane & 15;       // A/D row -> o ; B/D col -> b (same index)
  const int half = lane >> 4;       // K-half: lanes 16-31 carry K=2,3
  const int i0   = half * 2;

  const float* xb = x + (size_t)(bt * BT + mn) * (CIN * R_);   // x[b][i][r]
  const float* wc = w + (size_t)c * R_ * (CIN * O_);

  // logits = 0
  for (int k = t; k < BT * R_; k += 256) logits[k / R_][k % R_] = 0.0f;
  __syncthreads();

  const int sb = t & 15;            // stats: b
  const int sl = t >> 4;            // stats: r-slice (16 slices x 50)
  const int r0 = sl * (R_ / 16);

  for (int it = 0; it < 3; ++it) {
    // ---------- softmax stats over r, per b ----------
    float pm = -3.402823466e38f;
    for (int k = 0; k < R_ / 16; ++k) pm = fmaxf(pm, logits[sb][r0 + k]);
    pmax[sl][sb] = pm;
    __syncthreads();
    if (t < BT) {
      float m = -3.402823466e38f;
      for (int k = 0; k < 16; ++k) m = fmaxf(m, pmax[k][t]);
      mxl[t] = m;
    }
    __syncthreads();
    {
      float m = mxl[sb], ps = 0.0f;
      for (int k = 0; k < R_ / 16; ++k) ps += __expf(logits[sb][r0 + k] - m);
      psum[sl][sb] = ps;
    }
    __syncthreads();
    if (t < BT) {
      float s = 0.0f;
      for (int k = 0; k < 16; ++k) s += psum[k][t];
      rsl[t] = 1.0f / s;
    }
    __syncthreads();

    // ---------- pass 1: s[o,b] = sum_r prob*priors (prob folded into B) ----
    const float mref = mxl[mn];
    const float rs   = rsl[mn];
    v8f acc;
    #pragma unroll
    for (int j = 0; j < 8; ++j) acc[j] = 0.0f;

    for (int r = wv * RPW; r < wv * RPW + RPW; ++r) {
      const float* wr = wc + r * (CIN * O_);
      __builtin_prefetch(wr + 2 * CIN * O_, 0, 1);   // global_prefetch_b8
      v2f a0, a1;
      a0.x = wr[(i0 + 0) * O_ + mn];                 // W[c,r,i,o], o=mn
      a0.y = wr[(i0 + 1) * O_ + mn];
      a1.x = wr[(i0 + 4) * O_ + mn];
      a1.y = wr[(i0 + 5) * O_ + mn];
      const float p = __expf(logits[mn][r] - mref) * rs;
      v2f b0, b1;                                    // x[b,i,r] * prob[b,r]
      b0.x = xb[(i0 + 0) * R_ + r] * p;
      b0.y = xb[(i0 + 1) * R_ + r] * p;
      b1.x = xb[(i0 + 4) * R_ + r] * p;
      b1.y = xb[(i0 + 5) * R_ + r] * p;
      acc = wmma_f32_16x16x4(a0, b0, acc);
      acc = wmma_f32_16x16x4(a1, b1, acc);
    }
    #pragma unroll
    for (int j = 0; j < 8; ++j) wacc[wv][mn][half * 8 + j] = acc[j];
    __syncthreads();

    // reduce waves -> s_lds[b][o]
    {
      const int b = t >> 4, o = t & 15;
      float s = 0.0f;
      #pragma unroll
      for (int wi = 0; wi < NWAVE; ++wi) s += wacc[wi][b][o];
      s_lds[b][o] = s;
    }
    __syncthreads();
    // squash over o -> v_lds
    if (t < BT) {
      float sn = 0.0f;
      #pragma unroll
      for (int o = 0; o < O_; ++o) { float sv = s_lds[t][o]; sn += sv * sv; }
      const float scale = sn / ((1.0f + sn) * sqrtf(sn));
      #pragma unroll
      for (int o = 0; o < O_; ++o) v_lds[t][o] = s_lds[t][o] * scale;
    }
    __syncthreads();

    // ---------- pass 2: delta[b,r] = priors . v  ;  logits += delta --------
    if (it < 2) {
      float vreg[8];
      #pragma unroll
      for (int j = 0; j < 8; ++j) vreg[j] = v_lds[mn][half * 8 + j];

      for (int r = wv * RPW; r < wv * RPW + RPW; ++r) {
        const float* wr = wc + r * (CIN * O_);
        v2f a0, a1, b0, b1;
        a0.x = wr[(i0 + 0) * O_ + mn];
        a0.y = wr[(i0 + 1) * O_ + mn];
        a1.x = wr[(i0 + 4) * O_ + mn];
        a1.y = wr[(i0 + 5) * O_ + mn];
        b0.x = xb[(i0 + 0) * R_ + r];
        b0.y = xb[(i0 + 1) * R_ + r];
        b1.x = xb[(i0 + 4) * R_ + r];
        b1.y = xb[(i0 + 5) * R_ + r];
        v8f pr;
        #pragma unroll
        for (int j = 0; j < 8; ++j) pr[j] = 0.0f;
        pr = wmma_f32_16x16x4(a0, b0, pr);
        pr = wmma_f32_16x16x4(a1, b1, pr);
        float d = 0.0f;
        #pragma unroll
        for (int j = 0; j < 8; ++j) d = fmaf(pr[j], vreg[j], d);
        // combine o=0..7 with o=8..15 halves: lanes 0-15 read lane+16 (and 0x1f, or 0x10)
        const int swz = __builtin_amdgcn_ds_swizzle(__float_as_int(d), 0x021F);
        d += __int_as_float(swz);
        if (half == 0) logits[mn][r] += d;   // one lane per (b,r)
      }
      __syncthreads();
    }
  }

  // out[b][o][c]
  {
    const int bl = t >> 4, o = t & 15;
    out[(size_t)(bt * BT + bl) * (O_ * NC_) + (size_t)o * NC_ + c] =
        v_lds[bl][o];
  }
}

extern "C" void kernel_launch(void* const* d_in, const int* in_sizes, int n_in,
                              void* d_out, int out_size, void* d_ws, size_t ws_size,
                              hipStream_t stream) {
  const float* x = (const float*)d_in[0];          // [64][8][800]
  const float* w = (const float*)d_in[1];          // [128][800][8][16]
  float* out = (float*)d_out;                      // [64][16][128]
  (void)in_sizes; (void)n_in; (void)out_size; (void)d_ws; (void)ws_size;
  dim3 grid(B_ / BT, NC_);
  capsule_route_kernel<<<grid, 256, 0, stream>>>(x, w, out);
}